// LIFCorrelation_75737453298145
// MI455X (gfx1250) — compile-verified
//
#include <hip/hip_runtime.h>
#include <hip/hip_bf16.h>

typedef float v2f __attribute__((ext_vector_type(2)));
typedef float v4f __attribute__((ext_vector_type(4)));
typedef float v8f __attribute__((ext_vector_type(8)));

#define Bn   32
#define Hn   512
#define Kdim 512      // IN == H == 512

// ---------------------------------------------------------------------------
// Kernel A: z_new / v_new  (elementwise over [B,H] = 16384)
//   v_dec = v + 0.1*(i - v);  z = (v_dec - 1 > 0);  v_new = (1-z)*v_dec
// ---------------------------------------------------------------------------
__global__ __launch_bounds__(256) void lif_zv_kernel(
    const float* __restrict__ v, const float* __restrict__ i,
    float* __restrict__ z_new, float* __restrict__ v_new)
{
    int t = blockIdx.x * 256 + threadIdx.x;          // exactly 16384 threads
    float vv = v[t];
    float vd = vv + 0.1f * (i[t] - vv);              // DT*TAU_MEM_INV = 0.1
    float zn = (vd - 1.0f) > 0.0f ? 1.0f : 0.0f;     // heaviside(v_dec - V_TH)
    z_new[t] = zn;
    v_new[t] = (1.0f - zn) * vd;                     // V_RESET = 0
}

// ---------------------------------------------------------------------------
// Kernel B: i_new = 0.8*i + input @ Win^T + z @ Wrec^T  via V_WMMA_F32_16X16X4_F32
//   M = batch (32 -> 2 tiles), N = h (512 -> 32 tiles), K = 512 + 512.
//   One wave per 16x16 output tile: 64 waves = 16 blocks x 4 waves.
//
//   A fragment (16x4 f32): lane<16 -> rows m0+lane, K = k,k+1 in v0,v1;
//                          lane>=16 -> K = k+2,k+3.  -> one float2 load/lane.
//   B fragment (4x16): B[k][n] = W[n0+n][k] (W row-major, stride 512);
//                      mirrored layout -> float2 load of W[n0+lane%16][k+2*(lane/16)].
//   C/D (16x16 f32): VGPR r, lanes 0-15 -> M=r, lanes 16-31 -> M=r+8; N=lane%16.
// ---------------------------------------------------------------------------
__global__ __launch_bounds__(128) void lif_inew_wmma_kernel(
    const float* __restrict__ input, const float* __restrict__ z,
    const float* __restrict__ i_in,
    const float* __restrict__ w_in,  const float* __restrict__ w_rec,
    float* __restrict__ i_new)
{
    const int wave = (blockIdx.x * 128 + threadIdx.x) >> 5;  // 0..63
    const int lane = threadIdx.x & 31;
    const int lr   = lane & 15;
    const int lh   = lane >> 4;          // 0 or 1 (K half)
    const int m0   = (wave & 1) << 4;    // 0 or 16 (batch tile)
    const int n0   = (wave >> 1) << 4;   // 0..496 (h tile)

    const float* aIn  = input + (m0 + lr) * Kdim + 2 * lh;
    const float* aZ   = z     + (m0 + lr) * Kdim + 2 * lh;
    const float* bIn  = w_in  + (n0 + lr) * Kdim + 2 * lh;
    const float* bRec = w_rec + (n0 + lr) * Kdim + 2 * lh;

    v8f c = {0.f, 0.f, 0.f, 0.f, 0.f, 0.f, 0.f, 0.f};

    #pragma unroll 8
    for (int k = 0; k < Kdim; k += 4) {
        v2f a = *(const v2f*)(aIn + k);
        v2f b = *(const v2f*)(bIn + k);
        c = __builtin_amdgcn_wmma_f32_16x16x4_f32(false, a, false, b,
                                                  (short)0, c, false, false);
    }
    #pragma unroll 8
    for (int k = 0; k < Kdim; k += 4) {
        v2f a = *(const v2f*)(aZ + k);
        v2f b = *(const v2f*)(bRec + k);
        c = __builtin_amdgcn_wmma_f32_16x16x4_f32(false, a, false, b,
                                                  (short)0, c, false, false);
    }

    #pragma unroll
    for (int r = 0; r < 8; ++r) {
        int row = m0 + r + 8 * lh;           // batch index
        int col = n0 + lr;                   // h index
        int idx = row * Hn + col;
        i_new[idx] = c[r] + 0.8f * i_in[idx];   // i_decayed = (1 - DT*TAU_SYN_INV)*i
    }
}

// ---------------------------------------------------------------------------
// Kernel C: STDP correlation sensor step, streaming over [B, P, H] (P = 512).
//   Each thread handles 4 consecutive post-indices (float4 = b128 traffic).
//   Big tensors are read-once / write-once -> non-temporal hints.
// ---------------------------------------------------------------------------
__global__ __launch_bounds__(256) void corr_step_kernel(
    const float* __restrict__ pre_mat,   // [B, P]   (spikes, 0/1)
    const float* __restrict__ z_new,     // [B, H]   (post spikes)
    const float* __restrict__ pp_in,     // [B, P, H]
    const float* __restrict__ corr_in,   // [B, P, H]
    const float* __restrict__ anti_in,   // [B, P, H]
    float* __restrict__ pp_out,
    float* __restrict__ corr_out,
    float* __restrict__ anti_out)
{
    const int t = blockIdx.x * 256 + threadIdx.x;   // 2,097,152 threads
    const int e = t << 2;                           // flat element idx (mult of 4)
    const int b = e >> 18;                          // / (512*512)
    const int p = (e >> 9) & 511;
    const int j = e & 511;

    const float pre = pre_mat[(b << 9) + p];                    // broadcast scalar
    const v4f  post = *(const v4f*)(z_new + (b << 9) + j);      // hot 64KB, cached

    v4f pp   = __builtin_nontemporal_load((const v4f*)(pp_in   + e));
    v4f corr = __builtin_nontemporal_load((const v4f*)(corr_in + e));
    v4f anti = __builtin_nontemporal_load((const v4f*)(anti_in + e));

    // decay = 1 - DT*TAU_C_INV = 1 - DT*TAU_AC_INV = 0.99 ; ETA_P = ETA_M = 1
    v4f corr_new = corr * 0.99f + post * pp;
    v4f anti_new = anti * 0.99f + pre * (1.0f - pp);
    v4f pp_new   = pre + (1.0f - pre) * (1.0f - post) * pp;

    __builtin_nontemporal_store(pp_new,   (v4f*)(pp_out   + e));
    __builtin_nontemporal_store(corr_new, (v4f*)(corr_out + e));
    __builtin_nontemporal_store(anti_new, (v4f*)(anti_out + e));
}

// ---------------------------------------------------------------------------
extern "C" void kernel_launch(void* const* d_in, const int* in_sizes, int n_in,
                              void* d_out, int out_size, void* d_ws, size_t ws_size,
                              hipStream_t stream)
{
    const float* input  = (const float*)d_in[0];   // [B, IN]
    const float* z      = (const float*)d_in[1];   // [B, H]
    const float* v      = (const float*)d_in[2];   // [B, H]
    const float* i_in   = (const float*)d_in[3];   // [B, H]
    const float* ic_pp  = (const float*)d_in[4];   // [B, IN, H]
    const float* ic_c   = (const float*)d_in[5];
    const float* ic_a   = (const float*)d_in[6];
    const float* rc_pp  = (const float*)d_in[7];   // [B, H, H]
    const float* rc_c   = (const float*)d_in[8];
    const float* rc_a   = (const float*)d_in[9];
    const float* w_in   = (const float*)d_in[10];  // [H, IN] row-major
    const float* w_rec  = (const float*)d_in[11];  // [H, H]  row-major

    float* out = (float*)d_out;
    const int    BH  = Bn * Hn;                    // 16384
    const long long SYN = (long long)Bn * Kdim * Hn;  // 8,388,608

    float* o_znew   = out;
    float* o_vnew   = out + BH;
    float* o_inew   = out + 2 * BH;
    float* o_ic_pp  = out + 3 * BH;
    float* o_ic_c   = o_ic_pp + SYN;
    float* o_ic_a   = o_ic_c  + SYN;
    float* o_rc_pp  = o_ic_a  + SYN;
    float* o_rc_c   = o_rc_pp + SYN;
    float* o_rc_a   = o_rc_c  + SYN;

    // 1) z_new / v_new (z_new feeds the correlation kernels)
    lif_zv_kernel<<<BH / 256, 256, 0, stream>>>(v, i_in, o_znew, o_vnew);

    // 2) i_new via fp32 WMMA: 64 tiles = 16 blocks x 4 waves
    lif_inew_wmma_kernel<<<16, 128, 0, stream>>>(input, z, i_in, w_in, w_rec, o_inew);

    // 3) streaming correlation sensors: input->z_new and z->z_new
    const int nthr = (int)(SYN / 4);               // 2,097,152
    corr_step_kernel<<<nthr / 256, 256, 0, stream>>>(
        input, o_znew, ic_pp, ic_c, ic_a, o_ic_pp, o_ic_c, o_ic_a);
    corr_step_kernel<<<nthr / 256, 256, 0, stream>>>(
        z,     o_znew, rc_pp, rc_c, rc_a, o_rc_pp, o_rc_c, o_rc_a);
}